// MultiHeadAttention_3298534883398
// MI455X (gfx1250) — compile-verified
//
#include <hip/hip_runtime.h>
#include <hip/hip_bf16.h>
#include <math.h>

// Problem constants (match reference)
constexpr int B_  = 2;
constexpr int H_  = 8;
constexpr int LQ_ = 128;
constexpr int LK_ = 128;
constexpr int D_  = 512;   // num_hiddens
constexpr int DH_ = 64;    // per-head dim
constexpr int F_  = 512;   // additive-attention feature dim (== D)

typedef float v2f __attribute__((ext_vector_type(2)));
typedef float v8f __attribute__((ext_vector_type(8)));

#if __has_builtin(__builtin_amdgcn_tanhf)
#define TANH_F32(x) __builtin_amdgcn_tanhf(x)
#elif __has_builtin(__builtin_amdgcn_tanh_f32)
#define TANH_F32(x) __builtin_amdgcn_tanh_f32(x)
#else
#define TANH_F32(x) tanhf(x)
#endif

// ---------------------------------------------------------------------------
// One 16x16 output tile of C = A(M,K) @ W^T, where W is (N,K) row-major.
// Accumulates over full K with V_WMMA_F32_16X16X4_F32 (CDNA5 fp32 WMMA).
// ISA 7.12.2 layouts:
//   A 16x4 : lanes 0-15 -> M=lane, K={k0,k0+1}; lanes 16-31 -> K={k0+2,k0+3}
//   B 4x16 : lanes 0-15 -> N=lane, K={k0,k0+1}; lanes 16-31 -> K={k0+2,k0+3}
//   C/D    : vgpr r -> row m0+r (lanes 0-15) / m0+r+8 (lanes 16-31), col n0+lane%16
// ---------------------------------------------------------------------------
__device__ __forceinline__ v8f wmma_f32_tile(const float* __restrict__ A, int lda,
                                             const float* __restrict__ W, int ldw,
                                             int K, int m0, int n0, int lane)
{
  const int half = lane >> 4;   // 0 or 1
  const int l16  = lane & 15;
  const float* arow = A + (size_t)(m0 + l16) * lda + 2 * half;
  const float* wrow = W + (size_t)(n0 + l16) * ldw + 2 * half;
  v8f c = {};
  for (int k = 0; k < K; k += 4) {
    v2f a = *(const v2f*)(arow + k);   // A[m][k + 2*half .. +1]
    v2f b = *(const v2f*)(wrow + k);   // W[n][k + 2*half .. +1] == B^T fragment
    c = __builtin_amdgcn_wmma_f32_16x16x4_f32(
        /*neg_a=*/false, a, /*neg_b=*/false, b,
        /*c_mod=*/(short)0, c, /*reuse_a=*/false, /*reuse_b=*/false);
  }
  return c;
}

// ---------------------------------------------------------------------------
// Projection GEMM: out = X(M=256, K=512) @ W^T(512,512).
// split==1: scatter into head-split layout (B, H, L, DH)
// split==0: plain row-major (M, N)   [used for the final Wo projection]
// Grid: 64 blocks x 256 threads (8 waves, 1 tile/wave, 512 tiles total).
// ---------------------------------------------------------------------------
__global__ void __launch_bounds__(256)
proj_kernel(const float* __restrict__ X, const float* __restrict__ W,
            float* __restrict__ out, int split)
{
  const int M = B_ * LQ_;   // 256
  const int N = D_;         // 512
  const int K = D_;         // 512
  const int tid  = threadIdx.x;
  const int lane = tid & 31;
  const int wave = tid >> 5;
  const int tile = blockIdx.x * 8 + wave;        // 0..511
  const int tm = tile / (N / 16);
  const int tn = tile % (N / 16);
  const int m0 = tm * 16, n0 = tn * 16;

  v8f c = wmma_f32_tile(X, K, W, K, K, m0, n0, lane);

  const int l16 = lane & 15, half = lane >> 4;
  const int col = n0 + l16;
  for (int r = 0; r < 8; ++r) {
    const int row = m0 + r + 8 * half;
    if (split) {
      const int b = row / LQ_, l = row % LQ_;
      const int h = col / DH_, j = col % DH_;
      out[(((size_t)(b * H_ + h) * LQ_) + l) * DH_ + j] = c[r];
    } else {
      out[(size_t)row * N + col] = c[r];
    }
  }
}

// ---------------------------------------------------------------------------
// Additive-attention feature GEMM (batched over bh):
//   outf[bh] = Heads[bh](128, 64) @ Wa^T, Wa is (F=512, DH=64) row-major.
// Grid: (32, 16) x 256 threads; 256 tiles per bh, 8 waves/block.
// ---------------------------------------------------------------------------
__global__ void __launch_bounds__(256)
feat_kernel(const float* __restrict__ Hd, const float* __restrict__ Wa,
            float* __restrict__ outf)
{
  const int bh = blockIdx.y;
  const float* A = Hd + (size_t)bh * LQ_ * DH_;
  float* O = outf + (size_t)bh * LQ_ * F_;

  const int tid  = threadIdx.x;
  const int lane = tid & 31;
  const int wave = tid >> 5;
  const int tile = blockIdx.x * 8 + wave;        // 0..255
  const int tm = tile / (F_ / 16);
  const int tn = tile % (F_ / 16);
  const int m0 = tm * 16, n0 = tn * 16;

  v8f c = wmma_f32_tile(A, DH_, Wa, DH_, DH_, m0, n0, lane);

  const int l16 = lane & 15, half = lane >> 4;
  for (int r = 0; r < 8; ++r)
    O[(size_t)(m0 + r + 8 * half) * F_ + n0 + l16] = c[r];
}

// ---------------------------------------------------------------------------
// Fused: additive scores (tanh) -> masked softmax -> attn @ V -> ctx.
// One block (256 threads = 8 wave32) per (b, h, q) row.
//   scores[k] = sum_f wv_a[f] * tanh(qf[bh,q,f] + kf[bh,k,f])
// qf row + wv_a staged in LDS; wave-per-key, lanes stride F, shfl_xor reduce.
// ---------------------------------------------------------------------------
__global__ void __launch_bounds__(256)
score_softmax_av_kernel(const float* __restrict__ qf,   // (BH, LQ, F)
                        const float* __restrict__ kf,   // (BH, LK, F)
                        const float* __restrict__ vh,   // (BH, LK, DH)
                        const float* __restrict__ wv_a, // (F)
                        const int*   __restrict__ valid_lens, // (B)
                        float* __restrict__ ctx)        // (B, LQ, D) concat heads
{
  __shared__ float sq[F_];
  __shared__ float swv[F_];
  __shared__ float sp[LK_];
  __shared__ float spart[4 * DH_];
  __shared__ float swmax[8];
  __shared__ float swsum[8];

  const int bhq = blockIdx.x;       // 0 .. B*H*LQ-1
  const int qi  = bhq % LQ_;
  const int bh  = bhq / LQ_;
  const int b   = bh / H_;
  const int h   = bh % H_;
  const int tid  = threadIdx.x;
  const int lane = tid & 31;
  const int wave = tid >> 5;

  const float* qrow = qf + ((size_t)bh * LQ_ + qi) * F_;
  for (int f = tid; f < F_; f += 256) { sq[f] = qrow[f]; swv[f] = wv_a[f]; }
  __syncthreads();

  // ---- additive scores: one wave per key, 16 keys/wave ----
  const float* kbase = kf + (size_t)bh * LK_ * F_;
  for (int k = wave; k < LK_; k += 8) {
    const float* krow = kbase + (size_t)k * F_;
    float acc = 0.f;
    for (int f = lane; f < F_; f += 32)
      acc += swv[f] * TANH_F32(sq[f] + krow[f]);
    for (int off = 16; off > 0; off >>= 1)
      acc += __shfl_xor(acc, off, 32);
    if (lane == 0) sp[k] = acc;
  }
  __syncthreads();

  // ---- masked softmax over keys (threads 0..127 hold one key each) ----
  const int valid = valid_lens[b];
  float val = (tid < LK_) ? ((tid < valid) ? sp[tid] : -1.0e6f) : -3.0e38f;

  float m = val;
  for (int off = 16; off > 0; off >>= 1)
    m = fmaxf(m, __shfl_xor(m, off, 32));
  if (lane == 0) swmax[wave] = (wave < 4) ? m : -3.0e38f;
  __syncthreads();
  const float rowmax = fmaxf(fmaxf(swmax[0], swmax[1]), fmaxf(swmax[2], swmax[3]));

  float e = (tid < LK_) ? __expf(val - rowmax) : 0.f;  // masked: exp(~-1e6) -> 0
  float s = e;
  for (int off = 16; off > 0; off >>= 1)
    s += __shfl_xor(s, off, 32);
  if (lane == 0) swsum[wave] = (wave < 4) ? s : 0.f;
  __syncthreads();
  const float rowsum = swsum[0] + swsum[1] + swsum[2] + swsum[3];
  if (tid < LK_) sp[tid] = e / rowsum;
  __syncthreads();

  // ---- ctx[j] = sum_k attn[k] * V[k][j], 4 partial sums of 32 keys ----
  const int j = tid & 63;
  const int part = tid >> 6;                 // 0..3
  const float* vb = vh + (size_t)bh * LK_ * DH_;
  float acc = 0.f;
  for (int k = part * 32; k < part * 32 + 32; ++k)
    acc += sp[k] * vb[(size_t)k * DH_ + j];
  spart[part * DH_ + j] = acc;
  __syncthreads();

  if (tid < DH_) {
    const float o = spart[tid] + spart[DH_ + tid] + spart[2 * DH_ + tid] + spart[3 * DH_ + tid];
    ctx[((size_t)b * LQ_ + qi) * D_ + h * DH_ + tid] = o;
  }
}

// ---------------------------------------------------------------------------
extern "C" void kernel_launch(void* const* d_in, const int* in_sizes, int n_in,
                              void* d_out, int out_size, void* d_ws, size_t ws_size,
                              hipStream_t stream) {
  (void)in_sizes; (void)n_in; (void)out_size; (void)ws_size;

  const float* queries    = (const float*)d_in[0];   // (B, LQ, D)
  const float* keys       = (const float*)d_in[1];   // (B, LK, D)
  const float* values     = (const float*)d_in[2];   // (B, LK, D)
  const int*   valid_lens = (const int*)  d_in[3];   // (B)
  const float* Wq  = (const float*)d_in[4];          // (D, D)
  const float* Wk  = (const float*)d_in[5];
  const float* Wv  = (const float*)d_in[6];
  const float* Wo  = (const float*)d_in[7];
  const float* Wqa = (const float*)d_in[8];           // (D, dh)
  const float* Wka = (const float*)d_in[9];           // (D, dh)
  const float* wva = (const float*)d_in[10];          // (D,)

  // Workspace layout (floats): q,k,v head-split + qf,kf features + ctx
  float* ws  = (float*)d_ws;
  const size_t nQKV = (size_t)B_ * H_ * LQ_ * DH_;    // 131072 each
  const size_t nF   = (size_t)B_ * H_ * LQ_ * F_;     // 1048576 each
  float* q   = ws;
  float* k   = q  + nQKV;
  float* v   = k  + nQKV;
  float* qf  = v  + nQKV;
  float* kfb = qf + nF;
  float* ctx = kfb + nF;                              // total ~10.5 MB

  dim3 blk(256);
  proj_kernel<<<64, blk, 0, stream>>>(queries, Wq, q, 1);
  proj_kernel<<<64, blk, 0, stream>>>(keys,    Wk, k, 1);
  proj_kernel<<<64, blk, 0, stream>>>(values,  Wv, v, 1);

  feat_kernel<<<dim3(32, B_ * H_), blk, 0, stream>>>(q, Wqa, qf);
  feat_kernel<<<dim3(32, B_ * H_), blk, 0, stream>>>(k, Wka, kfb);

  score_softmax_av_kernel<<<B_ * H_ * LQ_, blk, 0, stream>>>(
      qf, kfb, v, wva, valid_lens, ctx);

  proj_kernel<<<64, blk, 0, stream>>>(ctx, Wo, (float*)d_out, 0);
}